// GraphEncoder_49134425866403
// MI455X (gfx1250) — compile-verified
//
#include <hip/hip_runtime.h>
#include <hip/hip_bf16.h>
#include <cstddef>

typedef float v2f __attribute__((ext_vector_type(2)));
typedef float v8f __attribute__((ext_vector_type(8)));

#define HEADS 4
#define CH 32
#define HID 128
#define NEG_SLOPE 0.2f
#define TM 128
#define TK 32

// ---------------------------------------------------------------------------
// GEMM: H = X[nrows x 128] * W[128 x 128] using fp32 WMMA 16x16x4.
// Block = 256 threads (8 wave32), computes a TM x 128 output tile.
// ---------------------------------------------------------------------------
__global__ __launch_bounds__(256)
void gat_gemm_f32(const float* __restrict__ X, const float* __restrict__ W,
                  float* __restrict__ Hout, int nrows) {
    __shared__ float xs[TM][TK + 1];   // +1 pad: avoid bank conflicts on column reads
    __shared__ float wt[TK][HID];

    const int tid  = threadIdx.x;
    const int wave = tid >> 5;
    const int lane = tid & 31;
    const int m    = lane & 15;   // row-in-tile (A) / col-in-tile (B, C)
    const int g    = lane >> 4;   // lane group: selects K=0,1 vs K=2,3 (A/B), M vs M+8 (C)
    const int row0 = blockIdx.x * TM;

    v8f acc[8] = {};

    for (int kc = 0; kc < HID; kc += TK) {
        // cooperative load of X tile (zero-fill past nrows)
        for (int idx = tid; idx < TM * TK; idx += 256) {
            int r = idx >> 5, c = idx & 31;
            int gr = row0 + r;
            xs[r][c] = (gr < nrows) ? X[(size_t)gr * HID + kc + c] : 0.0f;
        }
        // cooperative load of W tile (fully coalesced)
        for (int idx = tid; idx < TK * HID; idx += 256) {
            int kr = idx >> 7, c = idx & 127;
            wt[kr][c] = W[(size_t)(kc + kr) * HID + c];
        }
        __syncthreads();

        const int lr0 = wave * 16;
        #pragma unroll
        for (int kk = 0; kk < TK; kk += 4) {
            // A fragment (16x4 f32): lanes 0-15 hold K=kk,kk+1; lanes 16-31 K=kk+2,kk+3
            v2f a;
            a.x = xs[lr0 + m][kk + 2 * g];
            a.y = xs[lr0 + m][kk + 2 * g + 1];
            #pragma unroll
            for (int t = 0; t < 8; ++t) {
                // B fragment (4x16 f32): VGPR v, group g -> K = kk + v + 2g
                v2f b;
                b.x = wt[kk + 2 * g][t * 16 + m];
                b.y = wt[kk + 2 * g + 1][t * 16 + m];
                acc[t] = __builtin_amdgcn_wmma_f32_16x16x4_f32(
                    false, a, false, b, (short)0, acc[t], false, false);
            }
        }
        __syncthreads();
    }

    // C/D layout: VGPR i holds rows (wave*16 + 8g + i), col = t*16 + m
    #pragma unroll
    for (int t = 0; t < 8; ++t) {
        #pragma unroll
        for (int i = 0; i < 8; ++i) {
            int r = row0 + wave * 16 + g * 8 + i;
            if (r < nrows) Hout[(size_t)r * HID + t * 16 + m] = acc[t][i];
        }
    }
}

// ---------------------------------------------------------------------------
// Per-node attention logits: a_src[n,h] = dot(h[n,h,:], att_src[h,:]) etc.
// One thread per (node, head).
// ---------------------------------------------------------------------------
__global__ void gat_attn(const float* __restrict__ Hbuf,
                         const float* __restrict__ att_src,
                         const float* __restrict__ att_dst,
                         float* __restrict__ asrc, float* __restrict__ adst, int n) {
    int t = blockIdx.x * blockDim.x + threadIdx.x;
    if (t >= n * HEADS) return;
    int node = t >> 2, head = t & 3;
    const float* hp = Hbuf + (size_t)node * HID + head * CH;
    const float* sp = att_src + head * CH;
    const float* dp = att_dst + head * CH;
    float ss = 0.f, dd = 0.f;
    #pragma unroll
    for (int c = 0; c < CH; ++c) {
        float v = hp[c];
        ss += v * sp[c];
        dd += v * dp[c];
    }
    asrc[t] = ss;
    adst[t] = dd;
}

// ---------------------------------------------------------------------------
// Zero accumulator / sums; init segment-max keys (ordered-uint encoding).
// ---------------------------------------------------------------------------
__global__ void gat_init(float* __restrict__ acc, float* __restrict__ sbuf,
                         unsigned* __restrict__ mbuf, int n) {
    int t = blockIdx.x * blockDim.x + threadIdx.x;
    if (t < n * HID) acc[t] = 0.0f;
    if (t < n * HEADS) { sbuf[t] = 0.0f; mbuf[t] = 0u; }
}

// order-preserving float <-> uint maps (monotone wrt float ordering)
__device__ __forceinline__ unsigned f32_ord(float f) {
    unsigned u = __float_as_uint(f);
    return (u & 0x80000000u) ? ~u : (u | 0x80000000u);
}
__device__ __forceinline__ float ord_f32(unsigned u) {
    u = (u & 0x80000000u) ? (u & 0x7fffffffu) : ~u;
    return __uint_as_float(u);
}

// ---------------------------------------------------------------------------
// Edge pass 1: e = leaky_relu(a_src[src] + a_dst[dst]); segment max via
// integer atomicMax on order-preserving keys. One thread per edge.
// Edges [0,E) come from edge_index; [E, E+n) are self loops.
// ---------------------------------------------------------------------------
__global__ void gat_edge_logit(const int* __restrict__ ei, int E_, int n,
                               const float* __restrict__ asrc,
                               const float* __restrict__ adst,
                               float* __restrict__ ebuf, unsigned* __restrict__ mbuf) {
    int e = blockIdx.x * blockDim.x + threadIdx.x;
    int total = E_ + n;
    if (e >= total) return;
    int s, d;
    if (e < E_) { s = ei[e]; d = ei[e + E_]; } else { s = e - E_; d = s; }
    #pragma unroll
    for (int h = 0; h < HEADS; ++h) {
        float v = asrc[s * HEADS + h] + adst[d * HEADS + h];
        v = (v > 0.f) ? v : NEG_SLOPE * v;
        ebuf[(size_t)e * HEADS + h] = v;
        atomicMax(&mbuf[d * HEADS + h], f32_ord(v));
    }
}

// ---------------------------------------------------------------------------
// Edge pass 2: exp(e - m[dst]); segment sum via hardware fp32 atomic add.
// ---------------------------------------------------------------------------
__global__ void gat_edge_exp(const int* __restrict__ ei, int E_, int n,
                             const unsigned* __restrict__ mbuf,
                             float* __restrict__ ebuf, float* __restrict__ sbuf) {
    int e = blockIdx.x * blockDim.x + threadIdx.x;
    int total = E_ + n;
    if (e >= total) return;
    int d = (e < E_) ? ei[e + E_] : (e - E_);
    #pragma unroll
    for (int h = 0; h < HEADS; ++h) {
        float mv = ord_f32(mbuf[d * HEADS + h]);
        float x = __expf(ebuf[(size_t)e * HEADS + h] - mv);
        ebuf[(size_t)e * HEADS + h] = x;
        unsafeAtomicAdd(&sbuf[d * HEADS + h], x);
    }
}

// ---------------------------------------------------------------------------
// Edge pass 3 (dominant cost): one wave32 per edge. Lane l covers head=l>>3,
// 4 channels (float4). Gather h[src] (512B contiguous per edge), scale by
// alpha, scatter with global_atomic_add_f32.
// ---------------------------------------------------------------------------
__global__ __launch_bounds__(256)
void gat_scatter(const int* __restrict__ ei, int E_, int n,
                 const float* __restrict__ Hbuf, const float* __restrict__ ebuf,
                 const float* __restrict__ sbuf, float* __restrict__ acc) {
    int gt = blockIdx.x * blockDim.x + threadIdx.x;
    int e = gt >> 5;
    int lane = gt & 31;
    int total = E_ + n;
    if (e >= total) return;
    int s, d;
    if (e < E_) { s = ei[e]; d = ei[e + E_]; } else { s = e - E_; d = s; }
    int head = lane >> 3;
    int coff = head * CH + (lane & 7) * 4;
    float alpha = ebuf[(size_t)e * HEADS + head] /
                  (sbuf[d * HEADS + head] + 1e-16f);
    const float4 hv = *(const float4*)(Hbuf + (size_t)s * HID + coff);
    float* ap = acc + (size_t)d * HID + coff;
    unsafeAtomicAdd(ap + 0, hv.x * alpha);
    unsafeAtomicAdd(ap + 1, hv.y * alpha);
    unsafeAtomicAdd(ap + 2, hv.z * alpha);
    unsafeAtomicAdd(ap + 3, hv.w * alpha);
}

// ---------------------------------------------------------------------------
// out = relu(acc + bias)
// ---------------------------------------------------------------------------
__global__ void gat_finish(float* __restrict__ acc, const float* __restrict__ bias, int n) {
    int t = blockIdx.x * blockDim.x + threadIdx.x;
    if (t >= n * HID) return;
    float v = acc[t] + bias[t & (HID - 1)];
    acc[t] = (v > 0.f) ? v : 0.f;
}

// ---------------------------------------------------------------------------
extern "C" void kernel_launch(void* const* d_in, const int* in_sizes, int n_in,
                              void* d_out, int out_size, void* d_ws, size_t ws_size,
                              hipStream_t stream) {
    const int n  = in_sizes[0] / HID;     // 100000 (IN == HID == 128)
    const int E_ = in_sizes[1] / 2;       // 800000
    const int total = E_ + n;

    const float* x0 = (const float*)d_in[0];

    char* ws = (char*)d_ws;
    size_t off = 0;
    auto alloc = [&](size_t bytes) -> void* {
        void* p = ws + off;
        off += (bytes + 255) & ~(size_t)255;
        return p;
    };
    float*    hbuf = (float*)alloc((size_t)n * HID * sizeof(float));
    float*    xbuf = (float*)alloc((size_t)n * HID * sizeof(float));
    float*    asrc = (float*)alloc((size_t)n * HEADS * sizeof(float));
    float*    adst = (float*)alloc((size_t)n * HEADS * sizeof(float));
    float*    sbuf = (float*)alloc((size_t)n * HEADS * sizeof(float));
    unsigned* mbuf = (unsigned*)alloc((size_t)n * HEADS * sizeof(unsigned));
    float*    ebuf = (float*)alloc((size_t)total * HEADS * sizeof(float));

    for (int l = 0; l < 3; ++l) {
        const int*   ei   = (const int*)d_in[1 + l * 5];
        const float* Wl   = (const float*)d_in[2 + l * 5];
        const float* at_s = (const float*)d_in[3 + l * 5];
        const float* at_d = (const float*)d_in[4 + l * 5];
        const float* bias = (const float*)d_in[5 + l * 5];

        const float* xin = (l == 0) ? x0 : xbuf;
        float* acc = (l == 2) ? (float*)d_out : xbuf;

        gat_gemm_f32<<<(n + TM - 1) / TM, 256, 0, stream>>>(xin, Wl, hbuf, n);
        gat_attn<<<(n * HEADS + 255) / 256, 256, 0, stream>>>(hbuf, at_s, at_d, asrc, adst, n);
        gat_init<<<((size_t)n * HID + 255) / 256, 256, 0, stream>>>(acc, sbuf, mbuf, n);
        gat_edge_logit<<<(total + 255) / 256, 256, 0, stream>>>(ei, E_, n, asrc, adst, ebuf, mbuf);
        gat_edge_exp<<<(total + 255) / 256, 256, 0, stream>>>(ei, E_, n, mbuf, ebuf, sbuf);
        gat_scatter<<<(((size_t)total * 32) + 255) / 256, 256, 0, stream>>>(
            ei, E_, n, hbuf, ebuf, sbuf, acc);
        gat_finish<<<((size_t)n * HID + 255) / 256, 256, 0, stream>>>(acc, bias, n);
    }
}